// PCQLinear_7808250544309
// MI455X (gfx1250) — compile-verified
//
#include <hip/hip_runtime.h>
#include <math.h>

#define Q_MAX   255.0f
#define SMOOTH  0.999f

#define B_ROWS  10240
#define IN_DIM  4096
#define OUT_DIM 4096
#define NCLUST  10
#define ROWS_PER_CLUSTER (B_ROWS / NCLUST)   // 1024

#define BM 256
#define BN 256
#define BK 16
#define LDT (BK + 4)    // 20 floats/row: 16B-aligned rows, conflict-free b64 frag loads

typedef __attribute__((ext_vector_type(2))) float v2f;
typedef __attribute__((ext_vector_type(8))) float v8f;

// ---- order-preserving float<->uint mapping (deterministic min/max atomics) ----
__device__ __forceinline__ unsigned ordf(float f) {
    unsigned u = __float_as_uint(f);
    return (u & 0x80000000u) ? ~u : (u | 0x80000000u);
}
__device__ __forceinline__ float unordf(unsigned u) {
    unsigned b = (u & 0x80000000u) ? (u & 0x7fffffffu) : ~u;
    return __uint_as_float(b);
}

__device__ __forceinline__ float fakeq(float v, float s, float z) {
    return (fminf(fmaxf(rintf(v / s + z), 0.0f), Q_MAX) - z) * s;
}

// ---------------- K0: init atomic slots ----------------
__global__ void init_ws_kernel(unsigned* wmin, unsigned* wmax,
                               unsigned* cmin, unsigned* cmax) {
    int t = threadIdx.x;
    if (t == 0) { *wmin = 0xFFFFFFFFu; *wmax = 0u; }
    if (t < NCLUST) { cmin[t] = 0xFFFFFFFFu; cmax[t] = 0u; }
}

// ---------------- K1: global min/max of weight ----------------
__global__ void wminmax_kernel(const float* __restrict__ w,
                               unsigned* wmin, unsigned* wmax) {
    const int n4 = (OUT_DIM * IN_DIM) / 4;
    const float4* w4 = (const float4*)w;
    float lmin = 3.402823466e38f, lmax = -3.402823466e38f;
    for (int i = blockIdx.x * blockDim.x + threadIdx.x; i < n4;
         i += gridDim.x * blockDim.x) {
        float4 v = w4[i];
        lmin = fminf(lmin, fminf(fminf(v.x, v.y), fminf(v.z, v.w)));
        lmax = fmaxf(lmax, fmaxf(fmaxf(v.x, v.y), fmaxf(v.z, v.w)));
    }
    __shared__ float smin[256], smax[256];
    int t = threadIdx.x;
    smin[t] = lmin; smax[t] = lmax;
    __syncthreads();
    for (int s = 128; s > 0; s >>= 1) {
        if (t < s) {
            smin[t] = fminf(smin[t], smin[t + s]);
            smax[t] = fmaxf(smax[t], smax[t + s]);
        }
        __syncthreads();
    }
    if (t == 0) {
        atomicMin(wmin, ordf(smin[0]));
        atomicMax(wmax, ordf(smax[0]));
    }
}

// ---------------- K2: fake-quantize weight ----------------
__global__ void wquant_kernel(const float* __restrict__ w, float* __restrict__ wq,
                              const unsigned* wmin, const unsigned* wmax) {
    float mn = unordf(*wmin), mx = unordf(*wmax);
    float s = (mx - mn) / Q_MAX;
    float z = -rintf(mn / s);
    const int n4 = (OUT_DIM * IN_DIM) / 4;
    const float4* w4 = (const float4*)w;
    float4* q4 = (float4*)wq;
    for (int i = blockIdx.x * blockDim.x + threadIdx.x; i < n4;
         i += gridDim.x * blockDim.x) {
        float4 v = w4[i];
        v.x = fakeq(v.x, s, z); v.y = fakeq(v.y, s, z);
        v.z = fakeq(v.z, s, z); v.w = fakeq(v.w, s, z);
        q4[i] = v;
    }
}

// ---------------- K3: WMMA f32 GEMM  out = x @ wq^T + bias, + cluster min/max ----
__global__ __launch_bounds__(512)
void gemm_wmma_kernel(const float* __restrict__ x, const float* __restrict__ wq,
                      const float* __restrict__ bias, float* __restrict__ out,
                      unsigned* cmin, unsigned* cmax) {
    // K-contiguous tile rows: fragment = one aligned ds_load_b64, no register shuffles
    __shared__ float As[BM][LDT];   // As[m][k]
    __shared__ float Bs[BN][LDT];   // Bs[n][k]

    const int tid    = threadIdx.x;
    const int lane   = tid & 31;
    const int wid    = tid >> 5;          // 0..15
    const int mwave  = (wid >> 2) * 64;   // 4 waves along M
    const int nwave  = (wid & 3) * 64;    // 4 waves along N
    const int bm     = blockIdx.y * BM;
    const int bn     = blockIdx.x * BN;
    const int lane15 = lane & 15;
    const int khalf  = (lane >> 4) << 1;  // lanes 0-15 -> K {0,1}; lanes 16-31 -> K {2,3}

    v8f zeroAcc = {};
    v8f acc[4][4];
#pragma unroll
    for (int i = 0; i < 4; ++i)
#pragma unroll
        for (int j = 0; j < 4; ++j) acc[i][j] = zeroAcc;

    for (int k0 = 0; k0 < IN_DIM; k0 += BK) {
        // stage 256x16 tiles of x and wq into LDS with direct float4 stores
#pragma unroll
        for (int l = 0; l < 2; ++l) {
            int idx = tid + l * 512;        // 0..1023
            int r   = idx >> 2;             // 0..255
            int c4  = (idx & 3) << 2;       // 0,4,8,12
            float4 va = *(const float4*)(x  + (size_t)(bm + r) * IN_DIM + k0 + c4);
            *(float4*)&As[r][c4] = va;
            float4 vb = *(const float4*)(wq + (size_t)(bn + r) * IN_DIM + k0 + c4);
            *(float4*)&Bs[r][c4] = vb;
        }
        __syncthreads();

#pragma unroll
        for (int kk = 0; kk < BK; kk += 4) {
            v2f afrag[4], bfrag[4];
#pragma unroll
            for (int mt = 0; mt < 4; ++mt)
                afrag[mt] = *(const v2f*)&As[mwave + mt * 16 + lane15][kk + khalf];
#pragma unroll
            for (int nt = 0; nt < 4; ++nt)
                bfrag[nt] = *(const v2f*)&Bs[nwave + nt * 16 + lane15][kk + khalf];
#pragma unroll
            for (int mt = 0; mt < 4; ++mt)
#pragma unroll
                for (int nt = 0; nt < 4; ++nt)
                    acc[mt][nt] = __builtin_amdgcn_wmma_f32_16x16x4_f32(
                        false, afrag[mt], false, bfrag[nt],
                        (short)0, acc[mt][nt], false, false);
        }
        __syncthreads();
    }

    // epilogue: bias add, store fp32 out, track block min/max
    float lmin = 3.402823466e38f, lmax = -3.402823466e38f;
    const int rofs = (lane >> 4) << 3;   // C layout: VGPR j -> M=j (lanes 0-15), M=j+8 (16-31)
#pragma unroll
    for (int mt = 0; mt < 4; ++mt) {
#pragma unroll
        for (int nt = 0; nt < 4; ++nt) {
            int col = bn + nwave + nt * 16 + lane15;
            float bv = bias[col];
#pragma unroll
            for (int j = 0; j < 8; ++j) {
                int row = bm + mwave + mt * 16 + rofs + j;
                float v = acc[mt][nt][j] + bv;
                out[(size_t)row * OUT_DIM + col] = v;
                lmin = fminf(lmin, v);
                lmax = fmaxf(lmax, v);
            }
        }
    }
    __shared__ float smin[512], smax[512];
    smin[tid] = lmin; smax[tid] = lmax;
    __syncthreads();
    for (int s = 256; s > 0; s >>= 1) {
        if (tid < s) {
            smin[tid] = fminf(smin[tid], smin[tid + s]);
            smax[tid] = fmaxf(smax[tid], smax[tid + s]);
        }
        __syncthreads();
    }
    if (tid == 0) {
        int cid = bm / ROWS_PER_CLUSTER;   // 1024 % 256 == 0 -> block is cluster-pure
        atomicMin(&cmin[cid], ordf(smin[0]));
        atomicMax(&cmax[cid], ordf(smax[0]));
    }
}

// ---------------- K4: EMA ranges + per-cluster qparams ----------------
__global__ void finalize_kernel(const float* __restrict__ act_range,
                                const int* __restrict__ cluster_info,
                                const unsigned* __restrict__ cmin,
                                const unsigned* __restrict__ cmax,
                                float* __restrict__ new_range,
                                float* __restrict__ s3z3) {
    int t = threadIdx.x;
    if (t < NCLUST) {   // copy-through rows first (matches act_range.at[c].set)
        new_range[2 * t + 0] = act_range[2 * t + 0];
        new_range[2 * t + 1] = act_range[2 * t + 1];
    }
    __syncthreads();
    if (t < NCLUST) {
        int c = cluster_info[2 * t + 0];
        float mn = unordf(cmin[t]);
        float mx = unordf(cmax[t]);
        float nm = act_range[2 * c + 0] * SMOOTH + mn * (1.0f - SMOOTH);
        float nx = act_range[2 * c + 1] * SMOOTH + mx * (1.0f - SMOOTH);
        new_range[2 * c + 0] = nm;
        new_range[2 * c + 1] = nx;
        float s = (nx - nm) / Q_MAX;
        float z = -rintf(nm / s);
        s3z3[2 * t + 0] = s;
        s3z3[2 * t + 1] = z;
    }
}

// ---------------- K5: per-cluster fake-quantize out (in place) ----------------
__global__ void outquant_kernel(float* __restrict__ out,
                                const float* __restrict__ s3z3) {
    const size_t n4 = (size_t)B_ROWS * OUT_DIM / 4;
    float4* o4 = (float4*)out;
    size_t stride = (size_t)gridDim.x * blockDim.x;
    for (size_t i = (size_t)blockIdx.x * blockDim.x + threadIdx.x; i < n4; i += stride) {
        size_t row = (i * 4) / OUT_DIM;
        int k = (int)(row / ROWS_PER_CLUSTER);
        float s = s3z3[2 * k + 0];
        float z = s3z3[2 * k + 1];
        float4 v = o4[i];
        v.x = fakeq(v.x, s, z); v.y = fakeq(v.y, s, z);
        v.z = fakeq(v.z, s, z); v.w = fakeq(v.w, s, z);
        o4[i] = v;
    }
}

extern "C" void kernel_launch(void* const* d_in, const int* in_sizes, int n_in,
                              void* d_out, int out_size, void* d_ws, size_t ws_size,
                              hipStream_t stream) {
    const float* x            = (const float*)d_in[0];
    const float* w            = (const float*)d_in[1];
    const float* bias         = (const float*)d_in[2];
    const float* act_range    = (const float*)d_in[3];
    const int*   cluster_info = (const int*)d_in[4];

    float* out       = (float*)d_out;
    float* new_range = out + (size_t)B_ROWS * OUT_DIM;   // 20 floats at tail

    // workspace layout
    char* ws = (char*)d_ws;
    float*    wq   = (float*)ws;                                   // 64 MB
    size_t    off  = (size_t)OUT_DIM * IN_DIM * sizeof(float);
    unsigned* wmin = (unsigned*)(ws + off);
    unsigned* wmax = wmin + 1;
    unsigned* cmin = wmin + 2;            // NCLUST
    unsigned* cmax = cmin + NCLUST;       // NCLUST
    float*    s3z3 = (float*)(cmax + NCLUST);   // NCLUST * 2

    init_ws_kernel<<<1, 32, 0, stream>>>(wmin, wmax, cmin, cmax);
    wminmax_kernel<<<512, 256, 0, stream>>>(w, wmin, wmax);
    wquant_kernel<<<1024, 256, 0, stream>>>(w, wq, wmin, wmax);

    dim3 grid(OUT_DIM / BN, B_ROWS / BM);   // (16, 40)
    gemm_wmma_kernel<<<grid, 512, 0, stream>>>(x, wq, bias, out, cmin, cmax);

    finalize_kernel<<<1, 32, 0, stream>>>(act_range, cluster_info, cmin, cmax,
                                          new_range, s3z3);
    outquant_kernel<<<2048, 256, 0, stream>>>(out, s3z3);
}